// DFLoss_13116830122188
// MI455X (gfx1250) — compile-verified
//
#include <hip/hip_runtime.h>
#include <stdint.h>

// ---------------------------------------------------------------------------
// DFL loss, MI455X (gfx1250).
// Memory-bound (276 MB @ 23.3 TB/s ~ 12 us floor). Data path: TDM
// (tensor_load_to_lds) double-buffered global->LDS with pad-on-load so each
// 16-float group sits at an 18-dword stride (conflict-free across 32 lanes).
// ---------------------------------------------------------------------------

#define REG 16                         // DFL bins per side
#define GROUPS_PER_TILE 256            // (row,side) groups per TDM tile
#define TILE_ELEMS (GROUPS_PER_TILE * REG)       // 4096 floats per tile
#define GROUP_STRIDE_DW 18             // 16 data dwords + 2 pad dwords (TDM pad)
#define LDS_TILE_DW (GROUPS_PER_TILE * GROUP_STRIDE_DW)  // 4608 dwords
#define TILES_PER_BLOCK 8
#define BLOCK_THREADS 256

typedef unsigned int v4u __attribute__((ext_vector_type(4)));
typedef int          v8i __attribute__((ext_vector_type(8)));
typedef int          v4i __attribute__((ext_vector_type(4)));

// Build a Tensor DMA Descriptor (D#) per CDNA5 ISA ch.8 and issue
// TENSOR_LOAD_TO_LDS: one 1-line tile of 4096 f32, LDS padding 2 dwords every
// 16 dwords (pad_interval code 3 = 16 dw, pad_amount code 1 = 2 dw).
// This toolchain (clang-23 / therock-10.0) exposes the 6-arg builtin:
//   (v4u g0, v8i g1, v4i g2, v4i g3, v8i extra, i32 cpol)
__device__ __forceinline__ void tdm_load_tile(uint32_t lds_byte_addr,
                                              const float* gsrc) {
  uint64_t ga = (uint64_t)(uintptr_t)gsrc;
  v4u g0;
  g0.x = 1u;                                   // count=1 (valid), user mode
  g0.y = lds_byte_addr;                        // lds_addr (bytes)
  g0.z = (uint32_t)ga;                         // global_addr[31:0]
  g0.w = (uint32_t)((ga >> 32) & 0x01FFFFFFull) | (2u << 30); // addr[56:32]|type=2
  v8i g1;
  g1[0] = (int)((2u << 16)        // data_size = 4 bytes
              | (1u << 20)        // pad_enable
              | (3u << 22)        // pad_interval: every 16 dwords
              | (1u << 25));      // pad_amount: 2 dwords
  g1[1] = (int)((uint32_t)TILE_ELEMS << 16);   // tensor_dim0[15:0]
  g1[2] = (int)(1u << 16);                     // tensor_dim0[31:16]=0 | tensor_dim1=1
  g1[3] = (int)((uint32_t)TILE_ELEMS << 16);   // tensor_dim1 hi=0 | tile_dim0=4096
  g1[4] = 0;                                   // tile_dim1=0, tile_dim2=0 (unused)
  g1[5] = TILE_ELEMS;                          // tensor_dim0_stride
  g1[6] = 0;
  g1[7] = 0;
  v4i z4 = {0, 0, 0, 0};                       // groups 2/3: dims >=2 unused
  v8i z8 = {0, 0, 0, 0, 0, 0, 0, 0};           // extra descriptor words: unused
  __builtin_amdgcn_tensor_load_to_lds(g0, g1, z4, z4, z8, 0);
}

__global__ __launch_bounds__(BLOCK_THREADS) void dfl_main(
    const float* __restrict__ pred, const float* __restrict__ target,
    const float* __restrict__ weight, float* __restrict__ partials,
    int ntiles) {
  __shared__ float smem[2 * LDS_TILE_DW];            // double-buffered pred tiles
  __shared__ float wavesum[BLOCK_THREADS / 32];
  const int tid = threadIdx.x;
  // Flat LDS pointers carry the LDS byte offset in the low 32 bits (ISA 10.2).
  const uint32_t lds_base = (uint32_t)(uintptr_t)(&smem[0]);

  const int tile0 = blockIdx.x * TILES_PER_BLOCK;
  if (tile0 >= ntiles) return;
  float acc = 0.0f;

  if (tid < 32)  // wave 0 drives the TDM pipeline
    tdm_load_tile(lds_base, pred + (size_t)tile0 * TILE_ELEMS);

#pragma unroll 1
  for (int i = 0; i < TILES_PER_BLOCK; ++i) {
    const int tile = tile0 + i;
    if (i + 1 < TILES_PER_BLOCK) {
      if (tid < 32) {
        tdm_load_tile(lds_base + (uint32_t)(((i + 1) & 1) * (LDS_TILE_DW * 4)),
                      pred + (size_t)(tile + 1) * TILE_ELEMS);
        __builtin_amdgcn_s_wait_tensorcnt(1);  // tile i landed; i+1 in flight
      }
    } else {
      if (tid < 32) __builtin_amdgcn_s_wait_tensorcnt(0);
    }
    __syncthreads();  // publish tile i to all 8 waves

    // Thread tid owns group tid of this tile: 16 logits at 18-dword stride.
    const float* buf = &smem[(i & 1) * LDS_TILE_DW + GROUP_STRIDE_DW * tid];
    float v[REG];
#pragma unroll
    for (int j = 0; j < REG; ++j) v[j] = buf[j];

    float m = v[0];
#pragma unroll
    for (int j = 1; j < REG; ++j) m = fmaxf(m, v[j]);
    float se = 0.0f;
#pragma unroll
    for (int j = 0; j < REG; ++j) se += __expf(v[j] - m);
    const float lse = m + __logf(se);

    const int G = tile * GROUPS_PER_TILE + tid;      // global (row,side) index
    float t = target[G];
    t = fminf(fmaxf(t, 0.0f), (float)(REG - 1));
    const float fl = floorf(t);
    const int li = (int)fl;
    const int ri = min(li + 1, REG - 1);
    const float wr = t - fl;
    const float wl = 1.0f - wr;
    float ent = 0.0f;                                // xlogy semantics
    if (wl > 0.0f) ent += wl * __logf(wl);
    if (wr > 0.0f) ent += wr * __logf(wr);
    const float picked = wl * buf[li] + wr * buf[ri];
    const float kl = ent - picked + lse;
    acc += kl * weight[G >> 2];

    __syncthreads();  // done reading this buffer before TDM overwrites it
  }

  // wave32 tree reduce, then cross-wave via LDS (deterministic).
#pragma unroll
  for (int off = 16; off > 0; off >>= 1) acc += __shfl_down(acc, off, 32);
  if ((tid & 31) == 0) wavesum[tid >> 5] = acc;
  __syncthreads();
  if (tid == 0) {
    float s = 0.0f;
#pragma unroll
    for (int w = 0; w < BLOCK_THREADS / 32; ++w) s += wavesum[w];
    partials[blockIdx.x] = s;
  }
}

__global__ __launch_bounds__(256) void dfl_reduce(
    const float* __restrict__ partials, int n, float* __restrict__ out,
    float scale) {
  __shared__ float wavesum[8];
  const int tid = threadIdx.x;
  float a = 0.0f;
  for (int i = tid; i < n; i += 256) a += partials[i];
#pragma unroll
  for (int off = 16; off > 0; off >>= 1) a += __shfl_down(a, off, 32);
  if ((tid & 31) == 0) wavesum[tid >> 5] = a;
  __syncthreads();
  if (tid == 0) {
    float s = 0.0f;
#pragma unroll
    for (int w = 0; w < 8; ++w) s += wavesum[w];
    out[0] = s * scale;  // mean over 4*N group losses
  }
}

extern "C" void kernel_launch(void* const* d_in, const int* in_sizes, int n_in,
                              void* d_out, int out_size, void* d_ws,
                              size_t ws_size, hipStream_t stream) {
  const float* pred   = (const float*)d_in[0];   // [N, 64] f32
  const float* target = (const float*)d_in[1];   // [N, 4]  f32
  const float* weight = (const float*)d_in[2];   // [N, 1]  f32
  float* out = (float*)d_out;                    // scalar f32
  float* partials = (float*)d_ws;

  const int nrows   = in_sizes[0] / (4 * REG);           // N = 1,048,576
  const int ngroups = nrows * 4;                         // 4,194,304
  const int ntiles  = ngroups / GROUPS_PER_TILE;         // 16,384
  const int nblocks = ntiles / TILES_PER_BLOCK;          // 2,048

  dfl_main<<<nblocks, BLOCK_THREADS, 0, stream>>>(pred, target, weight,
                                                  partials, ntiles);
  const float scale = 1.0f / (float)ngroups;
  dfl_reduce<<<1, 256, 0, stream>>>(partials, nblocks, out, scale);
}